// LightGCN_1812476199038
// MI455X (gfx1250) — compile-verified
//
#include <hip/hip_runtime.h>
#include <stdint.h>

#define NUM_GROUPS 50000
#define NUM_ITEMS  100000
#define NUM_NODES  (NUM_GROUPS + NUM_ITEMS)
#define NUM_EDGES  2400000
#define EMBED      64
#define LAYERS     3
#define EPB        256   // edges per staged chunk
#define CHUNKS     5     // chunks per block: NUM_EDGES == 1875 * CHUNKS * EPB exactly

// ---------------------------------------------------------------------------
// init: x = concat(groups, items); accumulator (d_out, first 50k rows) = groups
// ---------------------------------------------------------------------------
__global__ __launch_bounds__(256) void lgcn_init(
    const float* __restrict__ groups, const float* __restrict__ items,
    float* __restrict__ x, float* __restrict__ out_acc) {
  const int i  = blockIdx.x * 256 + threadIdx.x;          // float4 index
  const int n4 = NUM_NODES  * EMBED / 4;
  const int g4 = NUM_GROUPS * EMBED / 4;
  if (i >= n4) return;
  float4 v = (i < g4) ? reinterpret_cast<const float4*>(groups)[i]
                      : reinterpret_cast<const float4*>(items)[i - g4];
  reinterpret_cast<float4*>(x)[i] = v;
  if (i < g4) reinterpret_cast<float4*>(out_acc)[i] = v;
}

// ---------------------------------------------------------------------------
// spmm: y[row[e]] += val[e] * x[col[e]]   (y pre-zeroed)
//
// Double-buffered pipeline over CHUNKS chunks of EPB edges:
//   stage(k+1) via 3x global_load_async_to_lds_b32   (ASYNCcnt += 3)
//   s_wait_asynccnt 3  -> drains chunk k (async loads complete in order)
//   barrier; consume chunk k (16 lanes/edge: b128 gather + 4 f32 atomics);
//   barrier (protect buffer reuse).
// ---------------------------------------------------------------------------
__global__ __launch_bounds__(256) void lgcn_spmm(
    const int* __restrict__ rows, const int* __restrict__ cols,
    const float* __restrict__ vals, const float* __restrict__ x,
    float* __restrict__ y) {
  __shared__ int   s_rows[2][EPB];
  __shared__ int   s_cols[2][EPB];
  __shared__ float s_vals[2][EPB];

  const int t    = threadIdx.x;
  const int base = blockIdx.x * (CHUNKS * EPB);

  auto stage = [&](int k) {
    const int b = k & 1;
    unsigned goff = (unsigned)(base + k * EPB + t) * 4u;   // byte offset (GVS mode)
    unsigned lr = (unsigned)(uintptr_t)(&s_rows[b][t]);    // low 32 bits = LDS addr
    unsigned lc = (unsigned)(uintptr_t)(&s_cols[b][t]);
    unsigned lv = (unsigned)(uintptr_t)(&s_vals[b][t]);
    asm volatile("global_load_async_to_lds_b32 %0, %1, %2 offset:0"
                 :: "v"(lr), "v"(goff), "s"(rows) : "memory");
    asm volatile("global_load_async_to_lds_b32 %0, %1, %2 offset:0"
                 :: "v"(lc), "v"(goff), "s"(cols) : "memory");
    asm volatile("global_load_async_to_lds_b32 %0, %1, %2 offset:0"
                 :: "v"(lv), "v"(goff), "s"(vals) : "memory");
  };

  stage(0);

  const int chunk = t & 15;   // which float4 of the 64-wide row
  const int slot  = t >> 4;   // which of 16 edges this pass

  for (int k = 0; k < CHUNKS; ++k) {
    if (k + 1 < CHUNKS) {
      stage(k + 1);                                        // overlap next chunk
      asm volatile("s_wait_asynccnt 3" ::: "memory");      // drain chunk k only
    } else {
      asm volatile("s_wait_asynccnt 0" ::: "memory");
    }
    __syncthreads();                                       // chunk k visible to all waves

    const int b = k & 1;
    __builtin_prefetch(x + (size_t)s_cols[b][t] * EMBED, 0, 0);

#pragma unroll 4
    for (int it = 0; it < EPB / 16; ++it) {
      const int   le = it * 16 + slot;
      const int   r  = s_rows[b][le];
      const int   c  = s_cols[b][le];
      const float v  = s_vals[b][le];
      const float4 xv = reinterpret_cast<const float4*>(x + c * EMBED)[chunk];
      float* yp = y + r * EMBED + chunk * 4;
      atomicAdd(yp + 0, v * xv.x);   // global_atomic_add_f32 (no return)
      atomicAdd(yp + 1, v * xv.y);
      atomicAdd(yp + 2, v * xv.z);
      atomicAdd(yp + 3, v * xv.w);
    }
    __syncthreads();                                       // buffer b reusable
  }
}

// ---------------------------------------------------------------------------
// accumulate layer output into the group-row accumulator (lives in d_out)
// ---------------------------------------------------------------------------
__global__ __launch_bounds__(256) void lgcn_accum_groups(
    float* __restrict__ out_acc, const float* __restrict__ y) {
  const int i  = blockIdx.x * 256 + threadIdx.x;
  const int g4 = NUM_GROUPS * EMBED / 4;
  if (i >= g4) return;
  float4 a = reinterpret_cast<float4*>(out_acc)[i];
  const float4 b = reinterpret_cast<const float4*>(y)[i];
  a.x += b.x; a.y += b.y; a.z += b.z; a.w += b.w;
  reinterpret_cast<float4*>(out_acc)[i] = a;
}

// ---------------------------------------------------------------------------
// finalize: mean over (LAYERS + 1) = multiply by 0.25, in place in d_out
// ---------------------------------------------------------------------------
__global__ __launch_bounds__(256) void lgcn_finalize(float* __restrict__ out) {
  const int i  = blockIdx.x * 256 + threadIdx.x;
  const int g4 = NUM_GROUPS * EMBED / 4;
  if (i >= g4) return;
  float4 a = reinterpret_cast<float4*>(out)[i];
  a.x *= 0.25f; a.y *= 0.25f; a.z *= 0.25f; a.w *= 0.25f;
  reinterpret_cast<float4*>(out)[i] = a;
}

// ---------------------------------------------------------------------------
extern "C" void kernel_launch(void* const* d_in, const int* in_sizes, int n_in,
                              void* d_out, int out_size, void* d_ws, size_t ws_size,
                              hipStream_t stream) {
  const float* groups = (const float*)d_in[0];
  const float* items  = (const float*)d_in[1];
  const float* vals   = (const float*)d_in[2];
  const int*   rows   = (const int*)d_in[3];
  const int*   cols   = (const int*)d_in[4];
  float* out = (float*)d_out;

  const size_t nfl = (size_t)NUM_NODES * EMBED;     // 9.6M floats per buffer
  float* bufA = (float*)d_ws;                       // ping
  float* bufB = bufA + nfl;                         // pong  (total 76.8 MB scratch)

  const int nodeBlocks  = (int)((nfl / 4 + 255) / 256);                        // 9375
  const int groupBlocks = (int)(((size_t)NUM_GROUPS * EMBED / 4 + 255) / 256); // 3125
  const int edgeBlocks  = NUM_EDGES / (CHUNKS * EPB);                          // 1875

  lgcn_init<<<nodeBlocks, 256, 0, stream>>>(groups, items, bufA, out);

  float* x = bufA;
  float* y = bufB;
  for (int l = 0; l < LAYERS; ++l) {
    hipMemsetAsync(y, 0, nfl * sizeof(float), stream);
    lgcn_spmm<<<edgeBlocks, EPB, 0, stream>>>(rows, cols, vals, x, y);
    lgcn_accum_groups<<<groupBlocks, 256, 0, stream>>>(out, y);
    float* tmp = x; x = y; y = tmp;
  }
  lgcn_finalize<<<groupBlocks, 256, 0, stream>>>(out);
}